// ChamferDistanceLoss_58377195487320
// MI455X (gfx1250) — compile-verified
//
#include <hip/hip_runtime.h>

typedef float v2f __attribute__((ext_vector_type(2)));
typedef float v8f __attribute__((ext_vector_type(8)));

#define CHUNK 1024            // target points staged in LDS per iteration
#define WAVES_PER_BLOCK 8
#define ROWS_PER_WAVE 16

// Each wave owns 16 "query" rows of P (B,N,3) and scans all M "reference"
// points of T (B,M,3) via V_WMMA_F32_16X16X4_F32 with tsq folded into K=4:
//   A row = [-2px, -2py, -2pz, 1],  B col = [tx, ty, tz, |t|^2]
//   => D = -2 p.t + |t|^2            (psq added AFTER the min; clamp after min:
//   min_m max(psq + D_m, 0) == max(psq + min_m D_m, 0), both maps monotone)
// B is pre-staged in LDS in the exact wave32 B-operand layout, so the inner
// loop is: ds_load_b64 + wmma + 8 v_min.
__global__ __launch_bounds__(256) void chamfer_min_kernel(
    const float* __restrict__ P, const float* __restrict__ T,
    float* __restrict__ partial, int N, int M)
{
    // per 16-point group g: float2[32]; lane l<16 -> {tx,ty}, l>=16 -> {tz,tsq}
    __shared__ float2 sB[CHUNK * 2];          // 16 KB

    const int lane  = threadIdx.x & 31;
    const int wave  = threadIdx.x >> 5;                  // 0..7
    const int b     = blockIdx.y;
    const int r0    = (blockIdx.x * WAVES_PER_BLOCK + wave) * ROWS_PER_WAVE;
    const int rlane = lane & 15;
    const int hi    = lane >> 4;

    // ---- A-matrix (16x4 f32 wave32 layout):
    // VGPR0: lanes0-15 K=0, lanes16-31 K=2 ; VGPR1: lanes0-15 K=1, lanes16-31 K=3
    const float* prow = P + ((size_t)b * N + (r0 + rlane)) * 3;
    const float px = prow[0], py = prow[1], pz = prow[2];
    const float psq = px * px + py * py + pz * pz;
    v2f a;
    a.x = hi ? (-2.0f * pz) : (-2.0f * px);
    a.y = hi ? 1.0f         : (-2.0f * py);

    float rmin[8];
#pragma unroll
    for (int i = 0; i < 8; ++i) rmin[i] = 3.4e38f;

    const float* Tb = T + (size_t)b * M * 3;

    for (int c0 = 0; c0 < M; c0 += CHUNK) {
        // ---- stage CHUNK reference points into LDS in B-operand layout ----
        for (int k = threadIdx.x; k < CHUNK; k += 256) {
            const float* t = Tb + (size_t)(c0 + k) * 3;
            const float x = t[0], y = t[1], z = t[2];
            const float tsq = x * x + y * y + z * z;
            const int g = k >> 4, col = k & 15;
            sB[g * 32 + col]      = make_float2(x, y);
            sB[g * 32 + 16 + col] = make_float2(z, tsq);
        }
        // prefetch next chunk while this one is consumed
        if (c0 + CHUNK < M) {
            const float* nxt = Tb + (size_t)(c0 + CHUNK) * 3 + threadIdx.x * 12;
            __builtin_prefetch(nxt, 0, 1);
        }
        __syncthreads();

#pragma unroll 4
        for (int g = 0; g < CHUNK / 16; ++g) {
            const float2 tb = sB[g * 32 + lane];         // one ds_load_b64
            v2f bb; bb.x = tb.x; bb.y = tb.y;
            v8f c = {};
            c = __builtin_amdgcn_wmma_f32_16x16x4_f32(
                    false, a, false, bb, (short)0, c, false, false);
            // C/D layout: VGPR i -> row = i + 8*hi, col = g*16 + (lane&15)
#pragma unroll
            for (int i = 0; i < 8; ++i) rmin[i] = fminf(rmin[i], c[i]);
        }
        __syncthreads();
    }

    // min across the 16 columns handled by each half-wave (xor stays in-half)
#pragma unroll
    for (int i = 0; i < 8; ++i) {
        float v = rmin[i];
        v = fminf(v, __shfl_xor(v, 1, 32));
        v = fminf(v, __shfl_xor(v, 2, 32));
        v = fminf(v, __shfl_xor(v, 4, 32));
        v = fminf(v, __shfl_xor(v, 8, 32));
        rmin[i] = v;
    }

    // add back the row-constant psq, clamp, and sum the 16 row-mins
    float s = 0.0f;
#pragma unroll
    for (int i = 0; i < 8; ++i) {
        const float pq = __shfl(psq, i + 8 * hi, 32);    // psq of row i + 8*hi
        s += fmaxf(pq + rmin[i], 0.0f);
    }
    s += __shfl_xor(s, 16, 32);          // combine both halves: 16-row sum

    if (lane == 0) {
        const int waveId = (b * gridDim.x + blockIdx.x) * WAVES_PER_BLOCK + wave;
        partial[waveId] = s;
    }
}

// Deterministic tree reduction of both directions' partial sums -> scalar.
__global__ __launch_bounds__(256) void chamfer_reduce_kernel(
    const float* __restrict__ partial, float* __restrict__ out,
    int nPerDir, float inv0, float inv1)
{
    __shared__ float s0s[256];
    __shared__ float s1s[256];
    float s0 = 0.0f, s1 = 0.0f;
    for (int i = threadIdx.x; i < nPerDir; i += 256) {
        s0 += partial[i];
        s1 += partial[nPerDir + i];
    }
    s0s[threadIdx.x] = s0;
    s1s[threadIdx.x] = s1;
    __syncthreads();
    for (int off = 128; off > 0; off >>= 1) {
        if ((int)threadIdx.x < off) {
            s0s[threadIdx.x] += s0s[threadIdx.x + off];
            s1s[threadIdx.x] += s1s[threadIdx.x + off];
        }
        __syncthreads();
    }
    if (threadIdx.x == 0) out[0] = s0s[0] * inv0 + s1s[0] * inv1;
}

extern "C" void kernel_launch(void* const* d_in, const int* in_sizes, int n_in,
                              void* d_out, int out_size, void* d_ws, size_t ws_size,
                              hipStream_t stream)
{
    const float* pred   = (const float*)d_in[0];   // (B, N, 3) fp32
    const float* target = (const float*)d_in[1];   // (B, M, 3) fp32

    const int B = 8;
    const int N = in_sizes[0] / (B * 3);           // 8192
    const int M = in_sizes[1] / (B * 3);           // 8192

    float* partial = (float*)d_ws;                 // 2 * nWaves floats
    float* out     = (float*)d_out;

    const int rowsPerBlock = WAVES_PER_BLOCK * ROWS_PER_WAVE;   // 128
    dim3 gridF(N / rowsPerBlock, B);               // pred -> target direction
    dim3 gridB(M / rowsPerBlock, B);               // target -> pred direction
    const int nWaves = (N / rowsPerBlock) * B * WAVES_PER_BLOCK;  // 4096

    chamfer_min_kernel<<<gridF, 256, 0, stream>>>(pred,   target, partial,          N, M);
    chamfer_min_kernel<<<gridB, 256, 0, stream>>>(target, pred,   partial + nWaves, M, N);

    chamfer_reduce_kernel<<<1, 256, 0, stream>>>(
        partial, out, nWaves, 1.0f / ((float)B * N), 1.0f / ((float)B * M));
}